// TrainOT_74569222193508
// MI455X (gfx1250) — compile-verified
//
#include <hip/hip_runtime.h>

typedef float v2f __attribute__((ext_vector_type(2)));
typedef float v8f __attribute__((ext_vector_type(8)));

#define GCN_DIM   128
#define REL_DIM   64
#define TRAIN_DIM 128
#define OUT_DIM   2
#define MT        5     // M-tiles per wave in the WMMA GEMM (N = 625*80, 80 = 16*MT)

// ---------------- degree / norm ----------------
__global__ void k_deg_init(float* deg, int n) {
    int i = blockIdx.x * blockDim.x + threadIdx.x;
    if (i < n) deg[i] = 1.0f;                 // self-loop contributes 1
}
__global__ void k_deg_accum(const int* __restrict__ dst, float* deg, int E) {
    int e = blockIdx.x * blockDim.x + threadIdx.x;
    if (e < E) atomicAdd(&deg[dst[e]], 1.0f);
}
__global__ void k_dinv(const float* __restrict__ deg, float* dinv, int n) {
    int i = blockIdx.x * blockDim.x + threadIdx.x;
    if (i < n) { float d = deg[i]; dinv[i] = d > 0.f ? rsqrtf(d) : 0.f; }
}

// ---------------- layer-1 projection: [N,3]@[3,128] ----------------
__global__ void k_xw1(const float* __restrict__ feat, const float* __restrict__ W1,
                      float* __restrict__ out, int n) {
    int idx = blockIdx.x * blockDim.x + threadIdx.x;
    if (idx >= n * GCN_DIM) return;
    int node = idx >> 7, c = idx & (GCN_DIM - 1);
    const float* f = feat + (long)node * 3;
    out[idx] = f[0] * W1[c] + f[1] * W1[GCN_DIM + c] + f[2] * W1[2 * GCN_DIM + c];
}

// ---------------- fp32 WMMA GEMM: C[rows,dout] = A[rows,K] @ B[K,dout] (+bias) ----
// One wave per (MT x 16) x 16 C-strip: B fragment loaded once per K-step and
// reused across MT back-to-back WMMAs. rows % (16*MT) == 0, dout % 16 == 0.
__global__ void k_gemm_wmma(const float* __restrict__ A, const float* __restrict__ B,
                            const float* __restrict__ bias, float* __restrict__ C,
                            int K, int dout) {
    int lane   = threadIdx.x;                 // 0..31, wave32
    int tilesN = dout >> 4;
    int tile   = blockIdx.x;
    int tg = tile / tilesN;                   // M-tile group (MT tiles)
    int tn = tile % tilesN;
    int l16  = lane & 15;
    int half = lane >> 4;                     // 0: K pair {0,1}, 1: K pair {2,3}
    int col  = tn * 16 + l16;                 // B/C column for this lane
    long arow[MT];
    #pragma unroll
    for (int t = 0; t < MT; ++t)
        arow[t] = (long)((tg * MT + t) * 16 + l16) * K;
    v8f c[MT] = {};
    for (int k = 0; k < K; k += 4) {
        int ka = k + half * 2;
        v2f b;
        b.x = B[(long)ka * dout + col];
        b.y = B[(long)(ka + 1) * dout + col];
        #pragma unroll
        for (int t = 0; t < MT; ++t) {
            v2f a;
            a.x = A[arow[t] + ka];
            a.y = A[arow[t] + ka + 1];
            c[t] = __builtin_amdgcn_wmma_f32_16x16x4_f32(false, a, false, b,
                                                         (short)0, c[t], false, false);
        }
    }
    float bc = bias ? bias[col] : 0.f;
    #pragma unroll
    for (int t = 0; t < MT; ++t) {
        int rbase = (tg * MT + t) * 16 + half * 8;   // lanes 16-31 hold M = r+8
        #pragma unroll
        for (int r = 0; r < 8; ++r)
            C[(long)(rbase + r) * dout + col] = c[t][r] + bc;
    }
}

// ---------------- GCN aggregation ----------------
__global__ void k_self_init(const float* __restrict__ xw, const float* __restrict__ dinv,
                            float* __restrict__ acc, int n) {
    int idx = blockIdx.x * blockDim.x + threadIdx.x;   // over n*32 float4s
    if (idx >= n * (GCN_DIM / 4)) return;
    int node = idx >> 5;
    float s = dinv[node]; s *= s;
    float4 v = ((const float4*)xw)[idx];
    v.x *= s; v.y *= s; v.z *= s; v.w *= s;
    ((float4*)acc)[idx] = v;
}

// one wave per edge; lane owns channels [4*lane, 4*lane+4)
__global__ void k_edge_scatter(const float* __restrict__ xw, float* acc,
                               const int* __restrict__ src, const int* __restrict__ dst,
                               const float* __restrict__ dinv, int E) {
    int wid  = (blockIdx.x * blockDim.x + threadIdx.x) >> 5;
    int lane = threadIdx.x & 31;
    if (wid >= E) return;
    int s = src[wid], d = dst[wid];
    float norm = dinv[s] * dinv[d];
    float4 m = ((const float4*)(xw + (long)s * GCN_DIM))[lane];
    float* a = acc + (long)d * GCN_DIM + lane * 4;
    atomicAdd(a + 0, m.x * norm);
    atomicAdd(a + 1, m.y * norm);
    atomicAdd(a + 2, m.z * norm);
    atomicAdd(a + 3, m.w * norm);
}

__global__ void k_bias_act(float* h, const float* __restrict__ b, int n, int relu) {
    int idx = blockIdx.x * blockDim.x + threadIdx.x;
    if (idx >= n * GCN_DIM) return;
    float v = h[idx] + b[idx & (GCN_DIM - 1)];
    if (relu) v = fmaxf(v, 0.f);
    h[idx] = v;
}

// ---------------- k = trainOT @ Wk + bk ; zero the reduction accumulator --------
__global__ void k_key_and_zero(const float* __restrict__ trainOT, const float* __restrict__ Wk,
                               const float* __restrict__ bk, float* kvec, float* summed) {
    int j = threadIdx.x;                      // 128 threads
    if (j < REL_DIM) {
        float s = bk[j];
        for (int i = 0; i < TRAIN_DIM; ++i) s += trainOT[i] * Wk[i * REL_DIM + j];
        kvec[j] = s;
    }
    summed[j] = 0.f;
}

// rel = sigmoid((q.k)/8); summed += rel * v   (one wave per node, grid-stride)
__global__ void k_rel_reduce(const float* __restrict__ q, const float* __restrict__ v,
                             const float* __restrict__ kvec, float* summed, int n) {
    int lane   = threadIdx.x & 31;
    int wid    = (blockIdx.x * blockDim.x + threadIdx.x) >> 5;
    int nwaves = (gridDim.x * blockDim.x) >> 5;
    float k0 = kvec[lane], k1 = kvec[32 + lane];
    float4 acc = {0.f, 0.f, 0.f, 0.f};
    for (int node = wid; node < n; node += nwaves) {
        float d = q[(long)node * REL_DIM + lane] * k0
                + q[(long)node * REL_DIM + 32 + lane] * k1;
        for (int off = 16; off; off >>= 1) d += __shfl_xor(d, off, 32);
        float rel = 1.f / (1.f + __expf(-d * 0.125f));   // scale = sqrt(64) = 8
        float4 vv = ((const float4*)(v + (long)node * TRAIN_DIM))[lane];
        acc.x += rel * vv.x; acc.y += rel * vv.y;
        acc.z += rel * vv.z; acc.w += rel * vv.w;
    }
    float* s = summed + lane * 4;
    atomicAdd(s + 0, acc.x); atomicAdd(s + 1, acc.y);
    atomicAdd(s + 2, acc.z); atomicAdd(s + 3, acc.w);
}

__global__ void k_newtrain(const float* __restrict__ trainOT, const float* __restrict__ summed,
                           float* nt, float* out_tail) {
    int i = threadIdx.x;                      // 128 threads
    float v = trainOT[i] + summed[i];
    nt[i] = v;
    out_tail[i] = v;
}

// speak[j] = sum_i nt[i] * Ws[i*M + j] + bs[j]
__global__ void k_speak(const float* __restrict__ nt, const float* __restrict__ Ws,
                        const float* __restrict__ bs, float* __restrict__ out, int M) {
    __shared__ float s_nt[TRAIN_DIM];
    if (threadIdx.x < TRAIN_DIM) s_nt[threadIdx.x] = nt[threadIdx.x];
    __syncthreads();
    int j = blockIdx.x * blockDim.x + threadIdx.x;
    if (j >= M) return;
    float s = bs[j];
    #pragma unroll 4
    for (int i = 0; i < TRAIN_DIM; ++i) s += s_nt[i] * Ws[(long)i * M + j];
    out[j] = s;
}

extern "C" void kernel_launch(void* const* d_in, const int* in_sizes, int n_in,
                              void* d_out, int out_size, void* d_ws, size_t ws_size,
                              hipStream_t stream) {
    const float* feat    = (const float*)d_in[0];
    const int*   edge    = (const int*)  d_in[1];
    const float* trainOT = (const float*)d_in[2];
    const float* W1      = (const float*)d_in[3];
    const float* b1      = (const float*)d_in[4];
    const float* W2      = (const float*)d_in[5];
    const float* b2      = (const float*)d_in[6];
    const float* Wq      = (const float*)d_in[7];
    const float* bq      = (const float*)d_in[8];
    const float* Wk      = (const float*)d_in[9];
    const float* bk      = (const float*)d_in[10];
    const float* Wv      = (const float*)d_in[11];
    const float* bv      = (const float*)d_in[12];
    const float* Ws      = (const float*)d_in[13];
    const float* bs      = (const float*)d_in[14];

    int N = in_sizes[0] / 3;          // 50000 (divisible by 16*MT = 80)
    int E = in_sizes[1] / 2;          // 800000
    int M = OUT_DIM * N;              // speak length
    const int* src = edge;
    const int* dst = edge + E;

    // workspace carve-up (floats)
    float* w = (float*)d_ws;
    float* buf0   = w; w += (size_t)N * GCN_DIM;   // xw1 / xw2 / v
    float* buf1   = w; w += (size_t)N * GCN_DIM;   // acc1/h1 / acc2/h2
    float* qbuf   = w; w += (size_t)N * REL_DIM;
    float* deg    = w; w += N;
    float* dinv   = w; w += N;
    float* kvec   = w; w += REL_DIM;
    float* summed = w; w += TRAIN_DIM;
    float* nt     = w; w += TRAIN_DIM;

    float* out = (float*)d_out;
    long nd = (long)N * GCN_DIM;
    int mgroups = N / (16 * MT);      // 625

    // normalization
    k_deg_init <<<(N + 255) / 256, 256, 0, stream>>>(deg, N);
    k_deg_accum<<<(E + 255) / 256, 256, 0, stream>>>(dst, deg, E);
    k_dinv     <<<(N + 255) / 256, 256, 0, stream>>>(deg, dinv, N);

    // GCN layer 1
    k_xw1         <<<(nd + 255) / 256, 256, 0, stream>>>(feat, W1, buf0, N);
    k_self_init   <<<(N * 32 + 255) / 256, 256, 0, stream>>>(buf0, dinv, buf1, N);
    k_edge_scatter<<<(E + 7) / 8, 256, 0, stream>>>(buf0, buf1, src, dst, dinv, E);
    k_bias_act    <<<(nd + 255) / 256, 256, 0, stream>>>(buf1, b1, N, 1);   // h1 = buf1

    // GCN layer 2 (dense part on WMMA)
    k_gemm_wmma   <<<mgroups * (GCN_DIM / 16), 32, 0, stream>>>(buf1, W2, nullptr, buf0, GCN_DIM, GCN_DIM);
    k_self_init   <<<(N * 32 + 255) / 256, 256, 0, stream>>>(buf0, dinv, buf1, N);
    k_edge_scatter<<<(E + 7) / 8, 256, 0, stream>>>(buf0, buf1, src, dst, dinv, E);
    k_bias_act    <<<(nd + 255) / 256, 256, 0, stream>>>(buf1, b2, N, 0);   // h2 = buf1

    // q, v projections on WMMA (bias fused)
    k_gemm_wmma<<<mgroups * (REL_DIM / 16),   32, 0, stream>>>(buf1, Wq, bq, qbuf, GCN_DIM, REL_DIM);
    k_gemm_wmma<<<mgroups * (TRAIN_DIM / 16), 32, 0, stream>>>(buf1, Wv, bv, buf0, GCN_DIM, TRAIN_DIM);

    // attention pooling
    k_key_and_zero<<<1, 128, 0, stream>>>(trainOT, Wk, bk, kvec, summed);
    k_rel_reduce  <<<512, 256, 0, stream>>>(qbuf, buf0, kvec, summed, N);
    k_newtrain    <<<1, 128, 0, stream>>>(trainOT, summed, nt, out + M);

    // speak -> d_out[0..M)
    k_speak<<<(M + 255) / 256, 256, 0, stream>>>(nt, Ws, bs, out, M);
}